// StaticPruner_47991964566142
// MI455X (gfx1250) — compile-verified
//
#include <hip/hip_runtime.h>
#include <cstddef>

typedef __attribute__((ext_vector_type(2))) float v2f;
typedef __attribute__((ext_vector_type(8))) float v8f;

#define BSZ 16
#define NH 12
#define SEQ 1024
#define HR (NH * SEQ)            // 12288 rows reduced per batch
#define NTILE (SEQ / 16)         // 64 column tiles per batch
#define NPAIR (BSZ * NTILE)      // 1024 (batch, tile) pairs
#define WPB 8                    // waves per block; one block per pair
#define ROWS_PER_WAVE (HR / WPB) // 1536
#define G16_PER_WAVE (ROWS_PER_WAVE / 16) // 96 sixteen-row groups

// ---------------------------------------------------------------------------
// Pass 1: column sums via V_WMMA_F32_16X16X4_F32 with an all-ones A matrix.
// Each wave owns one 16-column tile and 1536 rows. Each B-fragment slot is a
// VALU pre-sum of 4 rows, so one WMMA accumulates 16 rows into the column
// sums (D[m,n] = sum_k B[k,n] since A==1). Partials land in d_ws, one slice
// per wave id, summed deterministically in pass 2.
// ---------------------------------------------------------------------------
__global__ __launch_bounds__(256) void StaticPruner_reduce_kernel(
    const float* __restrict__ probes, float* __restrict__ part) {
  const int pair = blockIdx.x;         // 0..1023
  const int b    = pair >> 6;          // batch
  const int ct   = pair & (NTILE - 1); // column tile
  const int wave = threadIdx.x >> 5;   // 0..7
  const int lane = threadIdx.x & 31;
  const int kh   = lane >> 4;          // half-wave -> which K pair
  const int col  = (ct << 4) + (lane & 15);

  const float* __restrict__ p = probes
      + (size_t)b * HR * SEQ
      + (size_t)wave * ROWS_PER_WAVE * SEQ
      + col;
  const size_t kofs = (size_t)kh * 8 * SEQ; // kh=0 -> rows 0..7, kh=1 -> rows 8..15

  v2f a; a.x = 1.0f; a.y = 1.0f;   // all-ones A (layout-invariant)
  v8f c0 = {};
  v8f c1 = {};

  for (int it = 0; it < G16_PER_WAVE; it += 2) {
    const float* q0 = p + kofs;
    const float* q1 = q0 + (size_t)16 * SEQ;
    v2f f0, f1;
    // each fragment element = sum of 4 consecutive rows of this column
    f0.x = q0[0]       + q0[SEQ]     + q0[2 * SEQ] + q0[3 * SEQ];
    f0.y = q0[4 * SEQ] + q0[5 * SEQ] + q0[6 * SEQ] + q0[7 * SEQ];
    f1.x = q1[0]       + q1[SEQ]     + q1[2 * SEQ] + q1[3 * SEQ];
    f1.y = q1[4 * SEQ] + q1[5 * SEQ] + q1[6 * SEQ] + q1[7 * SEQ];
    p += (size_t)32 * SEQ;
    // D = ones(16x4) x B(4x16) + C  -> every D row holds the column sums
    c0 = __builtin_amdgcn_wmma_f32_16x16x4_f32(false, a, false, f0, (short)0,
                                               c0, false, false);
    c1 = __builtin_amdgcn_wmma_f32_16x16x4_f32(false, a, false, f1, (short)0,
                                               c1, false, false);
  }

  const float colsum = c0[0] + c1[0]; // VGPR0, lanes 0..15: (M=0, N=lane)
  if (lane < 16) {
    part[(size_t)wave * (BSZ * SEQ) + (size_t)b * SEQ + col] = colsum;
  }
}

// ---------------------------------------------------------------------------
// Pass 2: one block per batch (1024 threads). Deterministically combine the 8
// wave partials, compute length/sep/k, rank columns by score (stable
// descending order via index tie-break), and emit the pruned mask.
// ---------------------------------------------------------------------------
__global__ __launch_bounds__(1024) void StaticPruner_select_kernel(
    const float* __restrict__ part, const float* __restrict__ mask,
    float* __restrict__ out) {
  __shared__ float sc[SEQ];
  __shared__ float lensh;

  const int b = blockIdx.x;
  const int i = threadIdx.x;

  const float m = mask[(size_t)b * SEQ + i];

  if (i == 0) lensh = 0.0f;
  __syncthreads();
  // wave-local reduction of mask, then one LDS atomic per wave (exact: 0/1s)
  float v = m;
  for (int off = 16; off > 0; off >>= 1) v += __shfl_down(v, off, 32);
  if ((i & 31) == 0) atomicAdd(&lensh, v);
  __syncthreads();

  const float length = lensh;
  const int sep = (int)(length - 1.0f);
  const int k   = (int)(length * 0.4f); // (1 - PRUNING_RATIO)

  // fixed-order combine of the 8 partials (no atomics in this sum)
  float s = 0.0f;
  for (int w = 0; w < WPB; ++w)
    s += part[(size_t)w * (BSZ * SEQ) + (size_t)b * SEQ + i];

  const bool invalid = (i == 0) || (i == sep);
  sc[i] = invalid ? -__builtin_huge_valf() : s;
  __syncthreads();

  const float si = sc[i];
  int rank = 0;
  for (int j = 0; j < SEQ; ++j) {
    const float sj = sc[j];
    rank += (sj > si) || (sj == si && j < i);
  }

  bool keep = (rank < k) && !invalid;
  float o = keep ? 1.0f : 0.0f;
  if (i == 0 || i == sep) o = 1.0f;
  if (k <= 0) o = m;
  out[(size_t)b * SEQ + i] = o;
}

extern "C" void kernel_launch(void* const* d_in, const int* in_sizes, int n_in,
                              void* d_out, int out_size, void* d_ws,
                              size_t ws_size, hipStream_t stream) {
  const float* probes = (const float*)d_in[0]; // (16,12,1024,1024) f32
  const float* mask   = (const float*)d_in[1]; // (16,1,1,1024) f32
  float* out  = (float*)d_out;                 // (16,1024) f32
  float* part = (float*)d_ws;                  // 8 * 16 * 1024 floats = 512 KB

  StaticPruner_reduce_kernel<<<dim3(NPAIR), dim3(256), 0, stream>>>(probes,
                                                                    part);
  StaticPruner_select_kernel<<<dim3(BSZ), dim3(1024), 0, stream>>>(part, mask,
                                                                   out);
}